// EmbeddingShard_39307540693313
// MI455X (gfx1250) — compile-verified
//
#include <hip/hip_runtime.h>

// Shapes fixed by the reference model.
#define D_MODEL   4096
#define SEQ_LEN   2048
#define BLOCK     256
#define F4_PER_T  (D_MODEL / 4 / BLOCK)   // 4 float4 chunks per thread

typedef float f32x4 __attribute__((ext_vector_type(4)));

// out[b,s,:] = W[x[b,s],:] + b + pos_emb[s,:]
// One block per flattened token (B*S = 4096 blocks). Memory-bound gather+add:
// ~160 MiB HBM traffic -> ~7 us floor at 23.3 TB/s; AI ~ 0.2 flop/byte, so the
// async data-movement path (not WMMA) is the CDNA5 feature that matters here.
__global__ __launch_bounds__(BLOCK) void embed_gather_add_kernel(
    const int*   __restrict__ x,
    const float* __restrict__ W,
    const float* __restrict__ bias,
    const float* __restrict__ pos,
    float*       __restrict__ out)
{
  __shared__ float smemW[D_MODEL];          // 16 KiB staged gathered W row

  const int bs = blockIdx.x;                // flattened (batch, seq)
  const int s  = bs & (SEQ_LEN - 1);
  const int t  = threadIdx.x;

  const int token = x[bs];                  // uniform per block
  const float* Wrow = W   + (size_t)token * D_MODEL;
  const float* Prow = pos + (size_t)s     * D_MODEL;
  float*       Orow = out + (size_t)bs    * D_MODEL;

  // ---- Stage the HBM-random gathered W row into LDS via CDNA5 async copies.
  // GVS mode: saddr = uniform row base (SGPR pair), vaddr = per-lane byte
  // offset, vdst = LDS byte address. Tracked by ASYNCcnt, no VGPR return
  // dependency, so the wave issues all four copies and overlaps them with the
  // pos/bias loads below. W rows are read-once -> non-temporal, keep L2 for
  // pos_emb (reused across batch) and bias (reused by every block).
  #pragma unroll
  for (int k = 0; k < F4_PER_T; ++k) {
    const unsigned byteOff = (unsigned)(t + k * BLOCK) * 16u;
    const unsigned ldsAddr = (unsigned)(size_t)(&smemW[0]) + byteOff;
    asm volatile("global_load_async_to_lds_b128 %0, %1, %2 th:TH_LOAD_NT"
                 :
                 : "v"(ldsAddr), "v"(byteOff), "s"(Wrow)
                 : "memory");
  }

  // ---- Overlap: pull pos_emb row and bias (L2-friendly) while copies fly.
  f32x4 pv[F4_PER_T], bv[F4_PER_T];
  #pragma unroll
  for (int k = 0; k < F4_PER_T; ++k) {
    const int fi = (t + k * BLOCK) * 4;
    pv[k] = *(const f32x4*)(Prow + fi);
    bv[k] = *(const f32x4*)(bias + fi);
  }

  // Each thread reads back only the LDS region it staged itself, so the
  // per-wave async-counter wait is sufficient (no workgroup barrier needed).
  asm volatile("s_wait_asynccnt 0x0" ::: "memory");

  #pragma unroll
  for (int k = 0; k < F4_PER_T; ++k) {
    const int fi = (t + k * BLOCK) * 4;
    f32x4 wv = *(const f32x4*)(&smemW[fi]);        // ds_load_b128
    f32x4 o  = wv + bv[k] + pv[k];
    // Write-once 64 MiB result: non-temporal b128 store.
    __builtin_nontemporal_store(o, (f32x4*)(Orow + fi));
  }
}

extern "C" void kernel_launch(void* const* d_in, const int* in_sizes, int n_in,
                              void* d_out, int out_size, void* d_ws, size_t ws_size,
                              hipStream_t stream) {
  const int*   x    = (const int*)  d_in[0];  // [B, S] int32 token ids
  const float* W    = (const float*)d_in[1];  // [V, D]
  const float* bias = (const float*)d_in[2];  // [D]
  const float* pos  = (const float*)d_in[3];  // [S, D]
  float*       out  = (float*)d_out;          // [B, S, D]

  const int n_tok = in_sizes[0];              // B * S = 4096
  embed_gather_add_kernel<<<n_tok, BLOCK, 0, stream>>>(x, W, bias, pos, out);
}